// Attention_80315888435456
// MI455X (gfx1250) — compile-verified
//
#include <hip/hip_runtime.h>
#include <math.h>

// Sizes fixed by the reference.
#define B_      64
#define T_      2048
#define RNN_H_  1024
#define EMB_    512
#define ATT_    128
#define NFILT_  32
#define KW_     31
#define PAD_    15

typedef __attribute__((ext_vector_type(2))) float v2f;
typedef __attribute__((ext_vector_type(4))) float v4f;
typedef __attribute__((ext_vector_type(8))) float v8f;

// CDNA5 has a hardware V_TANH_F32 trans op; use it if clang exposes it.
static __device__ __forceinline__ float hw_tanhf(float x) {
#if __has_builtin(__builtin_amdgcn_tanhf)
  return __builtin_amdgcn_tanhf(x);
#elif __has_builtin(__builtin_amdgcn_tanh_f32)
  return __builtin_amdgcn_tanh_f32(x);
#else
  return tanhf(x);
#endif
}

// ---------------------------------------------------------------------------
// K0: after_query[b,a] = sum_h hidden[b,h] * W_dec[a,h]   (x @ W^T)
// 64 blocks x 128 threads; hidden row staged in LDS, W_dec rows via L2.
// ---------------------------------------------------------------------------
__global__ void k_after_query(const float* __restrict__ hidden,
                              const float* __restrict__ W_dec,
                              float* __restrict__ after_query) {
  __shared__ float hs[RNN_H_];
  const int b = blockIdx.x;
  for (int i = threadIdx.x; i < RNN_H_; i += blockDim.x)
    hs[i] = hidden[(size_t)b * RNN_H_ + i];
  __syncthreads();
  const int a = threadIdx.x;                 // 128 threads = ATT_
  const float* wr = W_dec + (size_t)a * RNN_H_;
  float acc = 0.f;
  for (int h = 0; h < RNN_H_; ++h) acc = fmaf(hs[h], wr[h], acc);
  after_query[b * ATT_ + a] = acc;
}

// ---------------------------------------------------------------------------
// K1: loc[b,t,f] = sum_c sum_k cum[b,c,t+k-PAD] * W_conv[f,c,k]
// One thread per (b,t); filters held in registers, weights in LDS (broadcast).
// Output layout [B,T,32] (filter-minor) == exactly the WMMA A-fragment order.
// ---------------------------------------------------------------------------
__global__ void k_conv_loc(const float* __restrict__ cum,
                           const float* __restrict__ W_conv,
                           float* __restrict__ loc) {
  __shared__ float wc[NFILT_ * 2 * KW_];     // 1984 floats
  for (int i = threadIdx.x; i < NFILT_ * 2 * KW_; i += blockDim.x)
    wc[i] = W_conv[i];
  __syncthreads();
  const int g = blockIdx.x * blockDim.x + threadIdx.x;   // flat (b,t)
  const int b = g / T_, t = g % T_;
  float acc[NFILT_];
#pragma unroll
  for (int f = 0; f < NFILT_; ++f) acc[f] = 0.f;
#pragma unroll
  for (int c = 0; c < 2; ++c) {
    const float* row = cum + ((size_t)b * 2 + c) * T_;
    for (int k = 0; k < KW_; ++k) {
      const int idx = t + k - PAD_;
      const float x = (idx >= 0 && idx < T_) ? row[idx] : 0.f;
#pragma unroll
      for (int f = 0; f < NFILT_; ++f)
        acc[f] = fmaf(x, wc[f * 2 * KW_ + c * KW_ + k], acc[f]);
    }
  }
  float* out = loc + (size_t)g * NFILT_;
#pragma unroll
  for (int f = 0; f < NFILT_; ++f) out[f] = acc[f];
}

// ---------------------------------------------------------------------------
// K2: fused  after_weights GEMM (WMMA f32 16x16x4) + q + memory_after
//            + tanh(V_TANH_F32) + dot(W_energy) + mask  ->  energies[b,t]
// Block = 256 threads (8 waves). Each wave owns 16 t-rows of one batch and
// loops the 8 att-tiles, chaining 8 K=4 WMMAs per tile (K total = 32).
// A 16x4 f32 frag: lanes 0-15 hold M=lane, VGPR0/1 = K {0,1}; lanes 16-31
// hold K {2,3}. B 4x16 frag mirrors it. C/D 16x16: VGPR v -> M = v + 8*half,
// N = lane%16.  EXEC is all-1s at every WMMA (no divergence before them).
// memory_after (64 MB, single-use) is loaded non-temporally to spare L2.
// ---------------------------------------------------------------------------
__global__ void k_energies_wmma(const float* __restrict__ loc,
                                const float* __restrict__ W_loc,
                                const float* __restrict__ after_query,
                                const float* __restrict__ memory_after,
                                const float* __restrict__ W_energy,
                                const unsigned char* __restrict__ mask,
                                float* __restrict__ energies) {
  const int b    = blockIdx.x >> 4;          // 16 blocks per batch
  const int tblk = blockIdx.x & 15;
  const int wave = threadIdx.x >> 5;
  const int lane = threadIdx.x & 31;
  const int half = lane >> 4;
  const int lr   = lane & 15;
  const int t0   = tblk * 128 + wave * 16;

  // A fragments are invariant across att-tiles: hoist all 8 K-chunks.
  const int arow = t0 + lr;
  const float* aBase = loc + ((size_t)b * T_ + arow) * NFILT_ + half * 2;
  v2f afr[8];
#pragma unroll
  for (int kk = 0; kk < 8; ++kk) {
    afr[kk].x = aBase[kk * 4 + 0];
    afr[kk].y = aBase[kk * 4 + 1];
  }

  float rs[8];
#pragma unroll
  for (int v = 0; v < 8; ++v) rs[v] = 0.f;

  for (int nt = 0; nt < 8; ++nt) {
    const int acol = nt * 16 + lr;           // att column owned by this lane
    const float* bBase = W_loc + (size_t)acol * NFILT_ + half * 2;
    v8f c = {};
#pragma unroll
    for (int kk = 0; kk < 8; ++kk) {
      v2f bv;
      bv.x = bBase[kk * 4 + 0];
      bv.y = bBase[kk * 4 + 1];
      c = __builtin_amdgcn_wmma_f32_16x16x4_f32(
          /*neg_a=*/false, afr[kk], /*neg_b=*/false, bv,
          /*c_mod=*/(short)0, c, /*reuse_a=*/false, /*reuse_b=*/false);
    }
    const float q  = after_query[b * ATT_ + acol];
    const float we = W_energy[acol];
#pragma unroll
    for (int v = 0; v < 8; ++v) {
      const int t = t0 + v + half * 8;
      const float ma = __builtin_nontemporal_load(
          memory_after + ((size_t)b * T_ + t) * ATT_ + acol);
      rs[v] += we * hw_tanhf(c[v] + q + ma);
    }
  }

  // Sum the 16 N-columns: xor offsets 1..8 stay inside each 16-lane half.
#pragma unroll
  for (int v = 0; v < 8; ++v) {
    float x = rs[v];
    x += __shfl_xor(x, 1, 32);
    x += __shfl_xor(x, 2, 32);
    x += __shfl_xor(x, 4, 32);
    x += __shfl_xor(x, 8, 32);
    rs[v] = x;
  }
  if (lr == 0) {                             // lanes 0 and 16 write 8 rows each
#pragma unroll
    for (int v = 0; v < 8; ++v) {
      const int t = t0 + v + half * 8;
      const float e = mask[(size_t)b * T_ + t] ? -INFINITY : rs[v];
      energies[(size_t)b * T_ + t] = e;
    }
  }
}

// ---------------------------------------------------------------------------
// K3: masked softmax over T per batch. One block (256 thr, 8 waves) per b;
// each thread holds 8 elements; wave shfl reduce + LDS across waves.
// ---------------------------------------------------------------------------
__global__ void k_softmax(const float* __restrict__ energies,
                          float* __restrict__ weights) {
  __shared__ float red[8];
  const int b = blockIdx.x, tid = threadIdx.x;
  const float* er = energies + (size_t)b * T_;
  float e[8];
#pragma unroll
  for (int i = 0; i < 8; ++i) e[i] = er[tid + i * 256];

  float m = -INFINITY;
#pragma unroll
  for (int i = 0; i < 8; ++i) m = fmaxf(m, e[i]);
  for (int off = 16; off >= 1; off >>= 1) m = fmaxf(m, __shfl_xor(m, off, 32));
  if ((tid & 31) == 0) red[tid >> 5] = m;
  __syncthreads();
  float bm = -INFINITY;
#pragma unroll
  for (int w = 0; w < 8; ++w) bm = fmaxf(bm, red[w]);
  __syncthreads();

  float s = 0.f;
#pragma unroll
  for (int i = 0; i < 8; ++i) { e[i] = __expf(e[i] - bm); s += e[i]; }
  for (int off = 16; off >= 1; off >>= 1) s += __shfl_xor(s, off, 32);
  if ((tid & 31) == 0) red[tid >> 5] = s;
  __syncthreads();
  float bs = 0.f;
#pragma unroll
  for (int w = 0; w < 8; ++w) bs += red[w];

  const float inv = 1.f / bs;
  float* wr = weights + (size_t)b * T_;
#pragma unroll
  for (int i = 0; i < 8; ++i) wr[tid + i * 256] = e[i] * inv;
}

// ---------------------------------------------------------------------------
// K4a: context partials. The 256 MB `memory` stream sets the runtime floor,
// so move it with global_load_b128 (one v4f per lane; a half-block covers a
// full 2 KB row) and a non-temporal hint (single-use stream; 320 MB of
// single-use data > 192 MB L2). grid (4 t-chunks, B), block 256: t split
// 2-way in-block (LDS pair-reduce) and 4-way across blocks into deterministic
// workspace partials — no float atomics, so graph replays are bit-identical.
// ---------------------------------------------------------------------------
__global__ void k_context_part(const float* __restrict__ weights,
                               const float* __restrict__ memory,
                               float* __restrict__ part) {
  __shared__ float wsm[T_ / 4];              // this block's 512 weights
  __shared__ v4f red[128];
  const int b = blockIdx.y, ts = blockIdx.x, tid = threadIdx.x;
  const int tbase = ts * (T_ / 4);
  for (int i = tid; i < T_ / 4; i += 256)
    wsm[i] = weights[(size_t)b * T_ + tbase + i];
  __syncthreads();
  const int j = tid & 127, half = tid >> 7;  // j: which v4f of the 512-f row
  const v4f* mrow =
      (const v4f*)(memory + ((size_t)b * T_ + tbase) * EMB_) + j;
  v4f acc = {0.f, 0.f, 0.f, 0.f};
  for (int t = half; t < T_ / 4; t += 2) {
    const float w = wsm[t];
    const v4f m = __builtin_nontemporal_load(mrow + (size_t)t * (EMB_ / 4));
    acc.x = fmaf(w, m.x, acc.x);
    acc.y = fmaf(w, m.y, acc.y);
    acc.z = fmaf(w, m.z, acc.z);
    acc.w = fmaf(w, m.w, acc.w);
  }
  if (half) red[j] = acc;
  __syncthreads();
  if (!half) {
    v4f o = red[j];
    o += acc;
    ((v4f*)part)[((size_t)b * 4 + ts) * (EMB_ / 4) + j] = o;
  }
}

// K4b: fold the 4 t-chunk partials -> context[b, 0:512].
__global__ void k_context_reduce(const float* __restrict__ part,
                                 float* __restrict__ context) {
  const int b = blockIdx.x, j = threadIdx.x;   // 128 threads, one v4f each
  const v4f* p = (const v4f*)part + (size_t)b * 4 * (EMB_ / 4) + j;
  const v4f s = p[0] + p[EMB_ / 4] + p[2 * (EMB_ / 4)] + p[3 * (EMB_ / 4)];
  ((v4f*)(context + (size_t)b * EMB_))[j] = s;
}

// ---------------------------------------------------------------------------
// launch: ws = loc (16 MB) | after_query (32 KB) | energies (512 KB)
//              | context partials (512 KB)   (~17.0 MB total)
// Outputs: d_out = [context B*512 | weights B*T] (reference return order).
// ---------------------------------------------------------------------------
extern "C" void kernel_launch(void* const* d_in, const int* in_sizes, int n_in,
                              void* d_out, int out_size, void* d_ws, size_t ws_size,
                              hipStream_t stream) {
  const float* hidden       = (const float*)d_in[0];
  const float* memory       = (const float*)d_in[1];
  const float* memory_after = (const float*)d_in[2];
  const float* cum          = (const float*)d_in[3];
  const unsigned char* mask = (const unsigned char*)d_in[4]; // jax bool = 1B
  const float* W_dec        = (const float*)d_in[5];
  const float* W_conv       = (const float*)d_in[6];
  const float* W_loc        = (const float*)d_in[7];
  const float* W_energy     = (const float*)d_in[8];

  float* context = (float*)d_out;
  float* weights = (float*)d_out + (size_t)B_ * EMB_;

  char*  ws          = (char*)d_ws;
  float* loc         = (float*)ws;                                   // B*T*32
  float* after_query = (float*)(ws + (size_t)B_ * T_ * NFILT_ * 4);  // B*128
  float* energies    = after_query + B_ * ATT_;                      // B*T
  float* part        = energies + (size_t)B_ * T_;                   // B*4*512

  k_after_query<<<B_, ATT_, 0, stream>>>(hidden, W_dec, after_query);
  k_conv_loc<<<(B_ * T_) / 128, 128, 0, stream>>>(cum, W_conv, loc);
  k_energies_wmma<<<B_ * (T_ / 128), 256, 0, stream>>>(
      loc, W_loc, after_query, memory_after, W_energy, mask, energies);
  k_softmax<<<B_, 256, 0, stream>>>(energies, weights);
  dim3 g4(4, B_);
  k_context_part<<<g4, 256, 0, stream>>>(weights, memory, part);
  k_context_reduce<<<B_, 128, 0, stream>>>(part, context);
}